// GCNIILayer_21912923144342
// MI455X (gfx1250) — compile-verified
//
#include <hip/hip_runtime.h>
#include <hip/hip_bf16.h>

// GCNII layer for MI455X (gfx1250, wave32).
// Pipeline: edge scatter (atomic f32) -> column BN stats -> fused affine
// (BN*gamma+beta folded with alpha-residual) -> WMMA f32 GEMM 0.5*(h2 + h2@W^T)
// -> ReLU.  h/h2 live in d_out (in-place safe: each GEMM block reads only its
// own 16-row strip before writing it).

typedef float v2f __attribute__((ext_vector_type(2)));
typedef float v8f __attribute__((ext_vector_type(8)));

#define D 128
#define ALPHA 0.1f
#define BETA 0.5f
#define BN_EPS 1e-5f

// ---------------- K0: zero h (d_out) and stats ----------------
__global__ void k_zero(float* __restrict__ h, float* __restrict__ stats, int total4) {
    int i = blockIdx.x * blockDim.x + threadIdx.x;
    if (i < total4) {
        ((float4*)h)[i] = make_float4(0.f, 0.f, 0.f, 0.f);
    }
    if (i < 512) stats[i] = 0.f;  // sum[128], sumsq[128], a[128], b[128]
}

// ---------------- K1: edge-weighted scatter-add ----------------
// One wave (32 lanes) per edge; float4 gather per lane; 4 f32 atomics.
__global__ void __launch_bounds__(256) k_edge_scatter(
    const float* __restrict__ feat, const float* __restrict__ mask,
    const int* __restrict__ src, const int* __restrict__ dst,
    float* __restrict__ h, int E) {
    int gtid = blockIdx.x * blockDim.x + threadIdx.x;
    int e    = gtid >> 5;
    int lane = threadIdx.x & 31;
    if (e >= E) return;
    int s = src[e];
    int d = dst[e];
    float m = mask[e];
    const float4 f = *(const float4*)(feat + (size_t)s * D + lane * 4);
    float* hp = h + (size_t)d * D + lane * 4;
    atomicAdd(hp + 0, f.x * m);
    atomicAdd(hp + 1, f.y * m);
    atomicAdd(hp + 2, f.z * m);
    atomicAdd(hp + 3, f.w * m);
}

// ---------------- K2: per-column sum / sumsq ----------------
__global__ void __launch_bounds__(128) k_col_stats(
    const float* __restrict__ h, float* __restrict__ sum,
    float* __restrict__ sumsq, int N) {
    int c = threadIdx.x;  // 0..127, coalesced across columns
    float s = 0.f, s2 = 0.f;
    for (int r = blockIdx.x; r < N; r += gridDim.x) {
        float v = h[(size_t)r * D + c];
        s += v;
        s2 += v * v;
    }
    atomicAdd(&sum[c], s);
    atomicAdd(&sumsq[c], s2);
}

// ---------------- K2b: fold BN + alpha-blend into per-column affine ----------
// h2 = (1-ALPHA) * ((h-mean)*rstd*gamma + beta) + ALPHA*x0
//    = a[c]*h + b[c] + ALPHA*x0
__global__ void __launch_bounds__(128) k_coeffs(
    const float* __restrict__ sum, const float* __restrict__ sumsq,
    const float* __restrict__ gamma, const float* __restrict__ beta_bn,
    float* __restrict__ a, float* __restrict__ b, int N) {
    int c = threadIdx.x;
    float invN = 1.0f / (float)N;
    float mean = sum[c] * invN;
    float var  = sumsq[c] * invN - mean * mean;
    float rstd = rsqrtf(var + BN_EPS);
    float g    = gamma[c];
    a[c] = (1.0f - ALPHA) * g * rstd;
    b[c] = (1.0f - ALPHA) * (beta_bn[c] - mean * rstd * g);
}

// ---------------- K3: in-place affine (h -> h2) ----------------
__global__ void __launch_bounds__(256) k_affine(
    float* __restrict__ h, const float* __restrict__ x0,
    const float* __restrict__ a, const float* __restrict__ b, int total4) {
    int i = blockIdx.x * blockDim.x + threadIdx.x;
    if (i >= total4) return;
    int cb = (i * 4) & (D - 1);
    float4 hv = ((float4*)h)[i];
    float4 xv = ((const float4*)x0)[i];
    hv.x = a[cb + 0] * hv.x + b[cb + 0] + ALPHA * xv.x;
    hv.y = a[cb + 1] * hv.y + b[cb + 1] + ALPHA * xv.y;
    hv.z = a[cb + 2] * hv.z + b[cb + 2] + ALPHA * xv.z;
    hv.w = a[cb + 3] * hv.w + b[cb + 3] + ALPHA * xv.w;
    ((float4*)h)[i] = hv;
}

// ---------------- K4: WMMA fp32 GEMM + blend + ReLU ----------------
// out = relu(0.5*h2 + 0.5*(h2 @ W^T)); one wave per 16-row strip.
// A (16x4 f32): lanes 0-15 -> M=lane, v.x=K, v.y=K+1; lanes 16-31 same M, K+2/K+3.
// B (4x16 f32): lanes 0-15 -> N=lane, v.x=K, v.y=K+1; lanes 16-31 same N, K+2/K+3.
// C/D (16x16 f32): VGPR r -> M=r (lanes 0-15) / M=r+8 (lanes 16-31), N=lane&15.
__global__ void __launch_bounds__(32) k_gemm_blend_relu(
    const float* __restrict__ h2, const float* __restrict__ Wm,
    float* __restrict__ out) {
    const int row0  = blockIdx.x * 16;
    const int lane  = threadIdx.x & 31;
    const int mn    = lane & 15;          // M for A, N for B/D
    const int khalf = (lane >> 4) << 1;   // 0 or 2

    v8f acc[8] = {};
    const float* arow = h2 + (size_t)(row0 + mn) * D;

    #pragma unroll 4
    for (int k0 = 0; k0 < D; k0 += 4) {
        v2f afrag;
        afrag.x = arow[k0 + khalf + 0];
        afrag.y = arow[k0 + khalf + 1];
        #pragma unroll
        for (int j = 0; j < 8; ++j) {
            // B[k, n] = W^T[k, n] = W[n, k]
            const float* wrow = Wm + (size_t)(j * 16 + mn) * D;
            v2f bfrag;
            bfrag.x = wrow[k0 + khalf + 0];
            bfrag.y = wrow[k0 + khalf + 1];
            acc[j] = __builtin_amdgcn_wmma_f32_16x16x4_f32(
                false, afrag, false, bfrag, (short)0, acc[j], false, false);
        }
    }

    // Epilogue: 0.5*h2 + 0.5*acc, ReLU, in-place store (own strip only).
    const int mbase = row0 + ((lane >> 4) << 3);
    #pragma unroll
    for (int j = 0; j < 8; ++j) {
        const int col = j * 16 + mn;
        #pragma unroll
        for (int r = 0; r < 8; ++r) {
            const size_t idx = (size_t)(mbase + r) * D + col;
            float v = BETA * h2[idx] + BETA * acc[j][r];  // BETA = 0.5 both ways
            out[idx] = fmaxf(v, 0.0f);
        }
    }
}

extern "C" void kernel_launch(void* const* d_in, const int* in_sizes, int n_in,
                              void* d_out, int out_size, void* d_ws, size_t ws_size,
                              hipStream_t stream) {
    (void)n_in; (void)out_size; (void)ws_size;
    const float* feat   = (const float*)d_in[0];
    const float* x0     = (const float*)d_in[1];
    const float* mask   = (const float*)d_in[2];
    const float* Wm     = (const float*)d_in[3];
    const float* gamma  = (const float*)d_in[4];
    const float* betabn = (const float*)d_in[5];
    const int*   src    = (const int*)d_in[6];
    const int*   dst    = (const int*)d_in[7];
    float*       out    = (float*)d_out;

    const int N = in_sizes[0] / D;   // 50000
    const int E = in_sizes[6];       // 800000

    float* stats = (float*)d_ws;     // [sum:128 | sumsq:128 | a:128 | b:128]
    float* sum   = stats;
    float* sumsq = stats + 128;
    float* acoef = stats + 256;
    float* bcoef = stats + 384;

    const int total4 = (N * D) / 4;

    // K0: zero h (in d_out) + stats
    k_zero<<<(total4 + 255) / 256, 256, 0, stream>>>(out, stats, total4);

    // K1: edge scatter (one wave per edge)
    {
        long long threads = (long long)E * 32;
        int blocks = (int)((threads + 255) / 256);
        k_edge_scatter<<<blocks, 256, 0, stream>>>(feat, mask, src, dst, out, E);
    }

    // K2: column stats
    k_col_stats<<<512, 128, 0, stream>>>(out, sum, sumsq, N);

    // K2b: affine coefficients
    k_coeffs<<<1, 128, 0, stream>>>(sum, sumsq, gamma, betabn, acoef, bcoef, N);

    // K3: in-place h -> h2
    k_affine<<<(total4 + 255) / 256, 256, 0, stream>>>(out, x0, acoef, bcoef, total4);

    // K4: WMMA GEMM + blend + ReLU (N multiple of 16: 50000 = 3125*16)
    k_gemm_blend_relu<<<N / 16, 32, 0, stream>>>(out, Wm, out);
}